// AttentionDecoder_1_3959959847631
// MI455X (gfx1250) — compile-verified
//
#include <hip/hip_runtime.h>
#include <math.h>

typedef __attribute__((ext_vector_type(2))) float v2f;
typedef __attribute__((ext_vector_type(8))) float v8f;
typedef __attribute__((ext_vector_type(4))) unsigned int v4u;
typedef __attribute__((ext_vector_type(8))) int v8i;
typedef __attribute__((ext_vector_type(4))) int v4i;

#define B_     64
#define NK_    2048
#define HID_   1024
#define EMB_   512
#define KEY_   512
#define VOCAB_ 50257

#if __has_builtin(__builtin_amdgcn_tensor_load_to_lds) && __has_builtin(__builtin_amdgcn_s_wait_tensorcnt)
#define HAS_TDM 1
#else
#define HAS_TDM 0
#endif

// ---------------------------------------------------------------------------
// f32 WMMA primitive: D(16x16) = A(16x4) * B(4x16) + C
// ---------------------------------------------------------------------------
__device__ __forceinline__ v8f wmma4(v2f a, v2f b, v8f c) {
  return __builtin_amdgcn_wmma_f32_16x16x4_f32(
      /*neg_a=*/false, a, /*neg_b=*/false, b,
      /*c_mod=*/(short)0, c, /*reuse_a=*/false, /*reuse_b=*/false);
}

#if HAS_TDM
// ---------------------------------------------------------------------------
// Issue one TDM 2D tile load (D# group0 + group1 per CDNA5 ISA 8.3/8.4).
// data_size = 4 bytes. Optional LDS padding: pad_amount dwords-code every
// 2^(pad_interval+1) dwords. 6-arg builtin form (clang-23 / therock-10.0).
// ---------------------------------------------------------------------------
__device__ __forceinline__ void tdm_load_2d(unsigned lds_off, const void* gptr,
                                            unsigned tile_d0, unsigned tile_d1,
                                            unsigned long stride0,
                                            unsigned tensor_d0, unsigned tensor_d1,
                                            int pad_en, int pad_interval_code,
                                            int pad_amount_code) {
  const unsigned long ga = (unsigned long)gptr;
  v4u g0;
  g0.x = 1u;                                              // count=1, user mode
  g0.y = lds_off;                                         // LDS byte address
  g0.z = (unsigned)(ga & 0xffffffffu);                    // global_addr[31:0]
  g0.w = (unsigned)((ga >> 32) & 0x01ffffffu) | (2u << 30); // addr[56:32]|type=2
  v8i g1;
  g1[0] = (int)((2u << 16) |                              // data_size=4B
                ((unsigned)pad_en << 20) |
                ((unsigned)pad_interval_code << 22) |
                ((unsigned)pad_amount_code << 25));
  g1[1] = (int)((tensor_d0 & 0xffffu) << 16);             // atomic_addr=0 | dim0.lo
  g1[2] = (int)((tensor_d0 >> 16) | ((tensor_d1 & 0xffffu) << 16));
  g1[3] = (int)((tensor_d1 >> 16) | (tile_d0 << 16));
  g1[4] = (int)tile_d1;                                   // tile_dim2 = 0
  g1[5] = (int)(stride0 & 0xffffffffu);
  g1[6] = (int)((stride0 >> 32) & 0xffffu);               // stride1 = 0
  g1[7] = 0;
  const v4i z4 = {0, 0, 0, 0};
  const v8i z8 = {0, 0, 0, 0, 0, 0, 0, 0};
  __builtin_amdgcn_tensor_load_to_lds(g0, g1, z4, z4, z8, 0);
}
#endif

// ---------------------------------------------------------------------------
// M=64 column-stripe GEMM: C[64,N] = A[64,K] * op(B) + bias.
// One wave per 16-wide N stripe, 4 independent accumulators covering all 64
// rows -> op(B) streamed from HBM exactly once, accumulation chains broken.
// ---------------------------------------------------------------------------
template <bool TRANSB>
__global__ void gemm_m64_wmma(const float* __restrict__ A, int lda,
                              const float* __restrict__ B, int ldb,
                              const float* __restrict__ bias,
                              float* __restrict__ C, int ldc, int N, int K) {
  const int wave = threadIdx.x >> 5;
  const int lane = threadIdx.x & 31;
  const int half = lane >> 4;
  const int lm   = lane & 15;
  const int tilesN = (N + 15) >> 4;
  const int tn = blockIdx.x * (blockDim.x >> 5) + wave;
  if (tn >= tilesN) return;
  const int n0 = tn << 4;
  const int bcol  = n0 + lm;
  const int bcolc = (bcol < N) ? bcol : (N - 1);          // clamp tail loads

  v8f c[4] = {};
  #pragma unroll 2
  for (int k0 = 0; k0 < K; k0 += 4) {
    const int ka = k0 + 2 * half;
    v2f b;
    if (TRANSB) {                                         // op(B)(k,n)=B[n*ldb+k]
      b = *(const v2f*)(B + (long)bcolc * ldb + ka);
    } else {                                              // op(B)(k,n)=B[k*ldb+n]
      b.x = B[(long)ka * ldb + bcolc];
      b.y = B[(long)(ka + 1) * ldb + bcolc];
    }
    #pragma unroll
    for (int mt = 0; mt < 4; ++mt) {
      const v2f a = *(const v2f*)(A + (long)(mt * 16 + lm) * lda + ka);
      c[mt] = wmma4(a, b, c[mt]);
    }
  }
  if (bcol < N) {
    const float bv = bias ? bias[bcol] : 0.0f;
    #pragma unroll
    for (int mt = 0; mt < 4; ++mt)
      #pragma unroll
      for (int r = 0; r < 8; ++r)
        C[(long)(mt * 16 + half * 8 + r) * ldc + bcol] = c[mt][r] + bv;
  }
}

// ---------------------------------------------------------------------------
// Fused attention scores, block-tiled:
//   score[b,n] = sum_h tanh((ann@Wk)[b,n,h] + qw[b,h]) * v[h]
// Block tile 128 rows x 64 cols; K chunked by 32, A/B double-buffered in LDS
// via TDM (TENSORcnt) when available. 8 waves = 4 row-groups x 2 col-groups;
// each wave: 2x2 grid of 16x16 WMMA accumulators. Epilogue fused, lane-
// reduced with shuffles, combined across col-blocks with f32 global atomics.
// ---------------------------------------------------------------------------
#define MBLK 128
#define NBLK 64
#define KC   32
#define ASTR 34   // 32 + 2 pad dwords: bank-conflict-free, 8B-aligned rows

__global__ void attn_scores(const float* __restrict__ ann,   // (B,NK,KEY)
                            const float* __restrict__ Wk,    // (KEY,HID)
                            const float* __restrict__ qw,    // (B,HID) incl bk
                            const float* __restrict__ vat,   // (HID)
                            float* __restrict__ scores) {    // (B,NK) zeroed
  __shared__ float As[2][MBLK * ASTR];   // 2*128*34*4 = 34816 B
  __shared__ float Bs[2][KC * NBLK];     // 2*32*64*4  = 16384 B

  const int  cb  = blockIdx.x;                 // col block (16)
  const long g0  = (long)blockIdx.y * MBLK;    // first global row (strip in 1 b)
  const int  b   = (int)(g0 >> 11);            // g0 / NK_
  const int  n0  = cb * NBLK;
  const int  tid = threadIdx.x;
  const int wave = tid >> 5, lane = tid & 31, half = lane >> 4, lm = lane & 15;
  const int wm = (wave & 3) * 32;              // wave rows within block
  const int wn = (wave >> 2) * 32;             // wave cols within block

  const float* __restrict__ aBase = ann + g0 * KEY_;
  const float* __restrict__ bBase = Wk + n0;
  const int NC = KEY_ / KC;                    // 16 chunks

#if HAS_TDM
  auto issue = [&](int buf, int ch) {
    if (tid == 0) {
      tdm_load_2d((unsigned)(uintptr_t)&As[buf][0], aBase + ch * KC,
                  /*tile_d0=*/KC, /*tile_d1=*/MBLK, /*stride0=*/KEY_,
                  /*tensor_d0=*/KEY_, /*tensor_d1=*/MBLK,
                  /*pad_en=*/1, /*interval 2^5=32 dw*/ 4, /*amount 2 dw*/ 1);
      tdm_load_2d((unsigned)(uintptr_t)&Bs[buf][0], bBase + (long)ch * KC * HID_,
                  /*tile_d0=*/NBLK, /*tile_d1=*/KC, /*stride0=*/HID_,
                  /*tensor_d0=*/NBLK, /*tensor_d1=*/KC,
                  /*pad_en=*/0, 0, 0);
    }
  };
#else
  auto issue = [&](int buf, int ch) {
    const float* a = aBase + ch * KC;
    for (int i = tid; i < MBLK * KC; i += 256) {
      const int r = i >> 5, k = i & (KC - 1);
      As[buf][r * ASTR + k] = a[(long)r * KEY_ + k];
    }
    const float* bsrc = bBase + (long)ch * KC * HID_;
    for (int i = tid; i < KC * NBLK; i += 256) {
      const int k = i >> 6, n = i & (NBLK - 1);
      Bs[buf][k * NBLK + n] = bsrc[(long)k * HID_ + n];
    }
  };
#endif

  v8f c[2][2] = {};
  auto compute = [&](int buf) {
    #pragma unroll
    for (int k0 = 0; k0 < KC; k0 += 4) {
      const int ka = k0 + 2 * half;
      v2f a0, a1, b0, b1;
      a0.x = As[buf][(wm + lm) * ASTR + ka];
      a0.y = As[buf][(wm + lm) * ASTR + ka + 1];
      a1.x = As[buf][(wm + 16 + lm) * ASTR + ka];
      a1.y = As[buf][(wm + 16 + lm) * ASTR + ka + 1];
      b0.x = Bs[buf][ka * NBLK + wn + lm];
      b0.y = Bs[buf][(ka + 1) * NBLK + wn + lm];
      b1.x = Bs[buf][ka * NBLK + wn + 16 + lm];
      b1.y = Bs[buf][(ka + 1) * NBLK + wn + 16 + lm];
      c[0][0] = wmma4(a0, b0, c[0][0]);
      c[0][1] = wmma4(a0, b1, c[0][1]);
      c[1][0] = wmma4(a1, b0, c[1][0]);
      c[1][1] = wmma4(a1, b1, c[1][1]);
    }
  };

#if HAS_TDM
  issue(0, 0);
  for (int ch = 0; ch < NC; ++ch) {
    if (ch + 1 < NC) issue((ch + 1) & 1, ch + 1);
    if (tid == 0) {
      if (ch + 1 < NC) __builtin_amdgcn_s_wait_tensorcnt(2);   // buf[ch] ready
      else             __builtin_amdgcn_s_wait_tensorcnt(0);
    }
    __syncthreads();                 // chunk ch visible to all waves
    compute(ch & 1);
    __syncthreads();                 // done reading before next DMA overwrite
  }
#else
  for (int ch = 0; ch < NC; ++ch) {
    issue(ch & 1, ch);
    __syncthreads();
    compute(ch & 1);
    __syncthreads();
  }
#endif

  // Fused epilogue: tanh(c + qw) * v, reduce along the 16 h-lanes.
  float part[2][8];
  #pragma unroll
  for (int mt = 0; mt < 2; ++mt)
    #pragma unroll
    for (int r = 0; r < 8; ++r) part[mt][r] = 0.0f;

  #pragma unroll
  for (int nt = 0; nt < 2; ++nt) {
    const int   h  = n0 + wn + nt * 16 + lm;
    const float qv = qw[b * HID_ + h];
    const float vv = vat[h];
    #pragma unroll
    for (int mt = 0; mt < 2; ++mt)
      #pragma unroll
      for (int r = 0; r < 8; ++r)
        part[mt][r] += tanhf(c[mt][nt][r] + qv) * vv;
  }

  #pragma unroll
  for (int mt = 0; mt < 2; ++mt)
    #pragma unroll
    for (int r = 0; r < 8; ++r) {
      float p = part[mt][r];
      p += __shfl_xor(p, 1, 32);
      p += __shfl_xor(p, 2, 32);
      p += __shfl_xor(p, 4, 32);
      p += __shfl_xor(p, 8, 32);
      if (lm == 0)                       // one lane per (half, row)
        atomicAdd(&scores[g0 + wm + mt * 16 + half * 8 + r], p);
    }
}

// ---------------------------------------------------------------------------
__global__ void zero_f(float* __restrict__ p, int n) {
  const int i = blockIdx.x * 256 + threadIdx.x;
  if (i < n) p[i] = 0.0f;
}

// Softmax over NK per batch row; writes a in place + into output region.
__global__ void softmax_nk(float* __restrict__ scores, float* __restrict__ a_out) {
  const int b = blockIdx.x, tid = threadIdx.x;
  __shared__ float sd[256];
  float* __restrict__ s = scores + b * NK_;

  float m = -INFINITY;
  for (int i = tid; i < NK_; i += 256) m = fmaxf(m, s[i]);
  sd[tid] = m; __syncthreads();
  for (int st = 128; st > 0; st >>= 1) {
    if (tid < st) sd[tid] = fmaxf(sd[tid], sd[tid + st]);
    __syncthreads();
  }
  m = sd[0]; __syncthreads();

  float sum = 0.0f;
  for (int i = tid; i < NK_; i += 256) sum += __expf(s[i] - m);
  sd[tid] = sum; __syncthreads();
  for (int st = 128; st > 0; st >>= 1) {
    if (tid < st) sd[tid] += sd[tid + st];
    __syncthreads();
  }
  const float inv = 1.0f / sd[0];
  for (int i = tid; i < NK_; i += 256) {
    const float av = __expf(s[i] - m) * inv;
    s[i] = av;
    a_out[b * NK_ + i] = av;
  }
}

// context[b,k] = sum_n a[b,n] * ann[b,n,k]  (second streaming pass over ann)
__global__ void context_k(const float* __restrict__ a,
                          const float* __restrict__ ann,
                          float* __restrict__ ctx) {
  const int b = blockIdx.x >> 1;
  const int k = ((blockIdx.x & 1) << 8) + threadIdx.x;
  const float* __restrict__ ap = a + b * NK_;
  const float* __restrict__ Ab = ann + (long)b * NK_ * KEY_ + k;
  float acc = 0.0f;
  #pragma unroll 4
  for (int n = 0; n < NK_; ++n) acc += ap[n] * Ab[(long)n * KEY_];
  ctx[b * KEY_ + k] = acc;
}

// x = concat(emb_table[id[b]], context[b])
__global__ void build_x(const int* __restrict__ ids,
                        const float* __restrict__ emb,
                        const float* __restrict__ ctx,
                        float* __restrict__ x) {
  const int b = blockIdx.x, tid = threadIdx.x;
  const long id = (long)ids[b];
  for (int i = tid; i < EMB_; i += 256) {
    x[b * (EMB_ + KEY_) + i]        = emb[id * EMB_ + i];
    x[b * (EMB_ + KEY_) + EMB_ + i] = ctx[b * KEY_ + i];
  }
}

// GRU gate fusion: h_new = (1-z)*n + z*Q
__global__ void gru_k(const float* __restrict__ gi, const float* __restrict__ gh,
                      const float* __restrict__ Q,
                      float* __restrict__ h_ws, float* __restrict__ o_out) {
  const int idx = blockIdx.x * 256 + threadIdx.x;
  const int b = idx >> 10, h = idx & (HID_ - 1);
  const float* __restrict__ gib = gi + b * 3 * HID_;
  const float* __restrict__ ghb = gh + b * 3 * HID_;
  const float r = 1.0f / (1.0f + __expf(-(gib[h] + ghb[h])));
  const float z = 1.0f / (1.0f + __expf(-(gib[h + HID_] + ghb[h + HID_])));
  const float n = tanhf(gib[h + 2 * HID_] + r * ghb[h + 2 * HID_]);
  const float hv = (1.0f - z) * n + z * Q[idx];
  h_ws[idx] = hv;
  o_out[idx] = hv;
}

// In-place log_softmax over the vocab dimension of the logp output region.
__global__ void logsoftmax_v(float* __restrict__ logp) {
  const int b = blockIdx.x, tid = threadIdx.x;
  float* __restrict__ row = logp + (long)b * VOCAB_;
  __shared__ float sd[256];

  float m = -INFINITY;
  for (int i = tid; i < VOCAB_; i += 256) m = fmaxf(m, row[i]);
  sd[tid] = m; __syncthreads();
  for (int st = 128; st > 0; st >>= 1) {
    if (tid < st) sd[tid] = fmaxf(sd[tid], sd[tid + st]);
    __syncthreads();
  }
  m = sd[0]; __syncthreads();

  float sum = 0.0f;
  for (int i = tid; i < VOCAB_; i += 256) sum += __expf(row[i] - m);
  sd[tid] = sum; __syncthreads();
  for (int st = 128; st > 0; st >>= 1) {
    if (tid < st) sd[tid] += sd[tid + st];
    __syncthreads();
  }
  const float lse = m + logf(sd[0]);
  for (int i = tid; i < VOCAB_; i += 256) row[i] = row[i] - lse;
}

// ---------------------------------------------------------------------------
// Launcher
// ---------------------------------------------------------------------------
extern "C" void kernel_launch(void* const* d_in, const int* in_sizes, int n_in,
                              void* d_out, int out_size, void* d_ws, size_t ws_size,
                              hipStream_t stream) {
  const int*   ids    = (const int*)d_in[0];
  const float* hidden = (const float*)d_in[1];   // Q = hidden[0] (B,HID)
  const float* ann    = (const float*)d_in[2];
  const float* emb    = (const float*)d_in[3];
  const float* Wk     = (const float*)d_in[4];
  const float* Wq     = (const float*)d_in[5];
  const float* bk     = (const float*)d_in[6];
  const float* vat    = (const float*)d_in[7];
  const float* W_ih   = (const float*)d_in[8];
  const float* W_hh   = (const float*)d_in[9];
  const float* b_ih   = (const float*)d_in[10];
  const float* b_hh   = (const float*)d_in[11];
  const float* W_out  = (const float*)d_in[12];
  const float* b_out  = (const float*)d_in[13];

  float* ws = (float*)d_ws;
  float* ws_qw  = ws;                                    // B*HID
  float* ws_gh  = ws_qw + B_ * HID_;                     // B*3H
  float* ws_gi  = ws_gh + B_ * 3 * HID_;                 // B*3H
  float* ws_sc  = ws_gi + B_ * 3 * HID_;                 // B*NK (scores -> a)
  float* ws_ctx = ws_sc + B_ * NK_;                      // B*KEY
  float* ws_x   = ws_ctx + B_ * KEY_;                    // B*(EMB+KEY)
  float* ws_h   = ws_x + B_ * (EMB_ + KEY_);             // B*HID

  float* out      = (float*)d_out;
  float* out_logp = out;                                 // (1,B,VOCAB)
  float* out_o    = out + (long)B_ * VOCAB_;             // (1,B,HID)
  float* out_a    = out_o + (long)B_ * HID_;             // (B,NK)

  auto stripes = [](int N) { return (((N + 15) >> 4) + 7) / 8; };

  // 1) qw = Q@Wq + bk
  gemm_m64_wmma<false><<<stripes(HID_), 256, 0, stream>>>(
      hidden, HID_, Wq, HID_, bk, ws_qw, HID_, HID_, HID_);
  // 2) gh = Q@W_hh^T + b_hh
  gemm_m64_wmma<true><<<stripes(3 * HID_), 256, 0, stream>>>(
      hidden, HID_, W_hh, HID_, b_hh, ws_gh, 3 * HID_, 3 * HID_, HID_);
  // 3) fused annotations@Wk + tanh + v-dot -> scores (atomic across col blocks)
  zero_f<<<(B_ * NK_ + 255) / 256, 256, 0, stream>>>(ws_sc, B_ * NK_);
  {
    dim3 grid(HID_ / NBLK, (B_ * NK_) / MBLK);           // 16 x 1024
    attn_scores<<<grid, 256, 0, stream>>>(ann, Wk, ws_qw, vat, ws_sc);
  }
  // 4) softmax over NK -> a
  softmax_nk<<<B_, 256, 0, stream>>>(ws_sc, out_a);
  // 5) context = a . annotations
  context_k<<<B_ * 2, 256, 0, stream>>>(ws_sc, ann, ws_ctx);
  // 6) x = [emb, context]
  build_x<<<B_, 256, 0, stream>>>(ids, emb, ws_ctx, ws_x);
  // 7) gi = x@W_ih^T + b_ih
  gemm_m64_wmma<true><<<stripes(3 * HID_), 256, 0, stream>>>(
      ws_x, EMB_ + KEY_, W_ih, EMB_ + KEY_, b_ih, ws_gi, 3 * HID_, 3 * HID_,
      EMB_ + KEY_);
  // 8) GRU gates -> h_new
  gru_k<<<(B_ * HID_) / 256, 256, 0, stream>>>(ws_gi, ws_gh, hidden, ws_h, out_o);
  // 9) logits = h@W_out^T + b_out straight into logp region (W_out read once)
  gemm_m64_wmma<true><<<stripes(VOCAB_), 256, 0, stream>>>(
      ws_h, HID_, W_out, HID_, b_out, out_logp, VOCAB_, VOCAB_, HID_);
  // 10) in-place log_softmax over vocab
  logsoftmax_v<<<B_, 256, 0, stream>>>(out_logp);
}